// GAMSmooth_12807592476724
// MI455X (gfx1250) — compile-verified
//
#include <hip/hip_runtime.h>

// Problem constants from the reference
#define N_UNIQUE 10000
#define N_BASES  10
#define FILTERS  16
#define BATCH    32
#define SEQLEN   100000

typedef __attribute__((ext_vector_type(2))) float v2f;
typedef __attribute__((ext_vector_type(4))) float v4f;   // native vector: OK for nontemporal builtins
typedef __attribute__((ext_vector_type(8))) float v8f;

// ---------------------------------------------------------------------------
// Phase 1: shrunk[10000,16] = X_spline[10000,10] @ kernel[10,16] + bias
// One wave (32 threads) per 16-row tile, using V_WMMA_F32_16X16X4_F32.
// K=10 is padded to 12 with zeros (3 chained K=4 WMMAs).
// ---------------------------------------------------------------------------
__global__ __launch_bounds__(32)
void spline_shrink_wmma(const float* __restrict__ X,     // [N_UNIQUE, N_BASES]
                        const float* __restrict__ Kc,    // [N_BASES, FILTERS]
                        const float* __restrict__ bias,  // [FILTERS]
                        float* __restrict__ shrunk)      // [N_UNIQUE, FILTERS]
{
    const int tile = blockIdx.x;          // 625 tiles of 16 rows
    const int lane = threadIdx.x;         // 0..31, all active (EXEC all-1s)
    const int half = lane >> 4;           // 0 or 1 (K-pair selector)
    const int lmod = lane & 15;           // M for A, N for B/C/D
    const int rowA = tile * 16 + lmod;    // A-matrix row for this lane

    // Accumulator pre-loaded with bias: D[m][n] starts at bias[n].
    // C layout: vgpr r, lanes 0-15 -> (M=r, N=lane), lanes 16-31 -> (M=r+8, N=lane-16)
    const float bv = bias[lmod];
    v8f acc;
#pragma unroll
    for (int i = 0; i < 8; ++i) acc[i] = bv;

    // 3 WMMA steps covering K = 0..11 (K=10,11 masked to zero). Branch-free
    // masking keeps EXEC all-1s (required for WMMA).
#pragma unroll
    for (int kk = 0; kk < 3; ++kk) {
        const int k0 = kk * 4 + half * 2;            // this lane's K pair
        const int ka = (k0 + 0 < N_BASES) ? (k0 + 0) : (N_BASES - 1);
        const int kb = (k0 + 1 < N_BASES) ? (k0 + 1) : (N_BASES - 1);
        const float ma = (k0 + 0 < N_BASES) ? 1.0f : 0.0f;
        const float mb = (k0 + 1 < N_BASES) ? 1.0f : 0.0f;

        // A: 16x4 f32, lane holds row M=lmod, VGPR0/1 = K = half*2 (+1)
        v2f a;
        a[0] = X[rowA * N_BASES + ka] * ma;
        a[1] = X[rowA * N_BASES + kb] * mb;

        // B: 4x16 f32, lane holds col N=lmod, VGPR0/1 = K = half*2 (+1)
        v2f b;
        b[0] = Kc[ka * FILTERS + lmod] * ma;
        b[1] = Kc[kb * FILTERS + lmod] * mb;

        acc = __builtin_amdgcn_wmma_f32_16x16x4_f32(
            /*neg_a=*/false, a, /*neg_b=*/false, b,
            /*c_mod=*/(short)0, acc, /*reuse_a=*/false, /*reuse_b=*/false);
    }

    // Store D: vgpr r -> row M = r + half*8, col N = lmod
#pragma unroll
    for (int r = 0; r < 8; ++r) {
        const int m = tile * 16 + r + half * 8;
        shrunk[m * FILTERS + lmod] = acc[r];
    }
}

// ---------------------------------------------------------------------------
// Phase 2: out[e,:] = shrunk[idx[e],:]  (bias already folded into shrunk)
// One thread per 16B chunk; 4 consecutive lanes cover one element ->
// fully coalesced 512B/wave b128 stores. Non-temporal on the 205MB output
// stream and on idx so the 640KB shrunk table stays resident in L2.
// ---------------------------------------------------------------------------
__global__ __launch_bounds__(256)
void gather_scatter(const int* __restrict__ idx,      // [BATCH*SEQLEN]
                    const v4f* __restrict__ shrunk4,  // [N_UNIQUE*4]
                    v4f* __restrict__ out4,           // [BATCH*SEQLEN*4]
                    int total4)
{
    const int t = blockIdx.x * blockDim.x + threadIdx.x;
    if (t >= total4) return;
    const int e = t >> 2;                 // element index
    const int c = t & 3;                  // 16B chunk within the 16 filters
    const int id = __builtin_nontemporal_load(&idx[e]);
    const v4f v = shrunk4[id * 4 + c];    // temporal: keep table hot in L2
    __builtin_nontemporal_store(v, &out4[t]);
}

extern "C" void kernel_launch(void* const* d_in, const int* in_sizes, int n_in,
                              void* d_out, int out_size, void* d_ws, size_t ws_size,
                              hipStream_t stream) {
    // setup_inputs() dict order: idx, X_spline, kernel, bias
    const int*   idx  = (const int*)d_in[0];
    const float* X    = (const float*)d_in[1];
    const float* Kc   = (const float*)d_in[2];
    const float* bias = (const float*)d_in[3];
    float* out = (float*)d_out;

    // Workspace: shrunk table [N_UNIQUE, FILTERS] f32 = 640 KB
    float* shrunk = (float*)d_ws;

    // Phase 1: 625 one-wave blocks
    spline_shrink_wmma<<<N_UNIQUE / 16, 32, 0, stream>>>(X, Kc, bias, shrunk);

    // Phase 2: one thread per 16B chunk of output
    const int total4 = BATCH * SEQLEN * 4;            // 12,800,000
    const int threads = 256;
    const int blocks = (total4 + threads - 1) / threads;
    gather_scatter<<<blocks, threads, 0, stream>>>(
        idx, (const v4f*)shrunk, (v4f*)out, total4);
}